// Model_28750511079730
// MI455X (gfx1250) — compile-verified
//
#include <hip/hip_runtime.h>
#include <cstddef>

typedef float v2f __attribute__((ext_vector_type(2)));
typedef float v8f __attribute__((ext_vector_type(8)));

// Problem dims (fixed by the reference's setup_inputs)
constexpr int B_     = 64;
constexpr int IN_PN  = 20000;
constexpr int OUT_PN = 5000;
constexpr int M_     = 9;
constexpr int CIN    = 32;
constexpr int COUT   = 64;

// sqrt(1-RR) and sqrt(RR) with RR = 0.5
constexpr float SA = 0.70710678118654752f;
constexpr float SB = 0.70710678118654752f;

// LDS layout for the B (weights) tile of one neighbor m:
// packed k-pair rows: element (k, o) lives at  (k>>1)*STRIDE + 2*o + (k&1)  dwords.
// STRIDE = 172 dwords: transposing stores are conflict-free, fragment reads nearly so.
constexpr int LDSW_STRIDE = 172;
constexpr int LDSW_ROWS   = CIN / 2;   // 16 packed rows per m

__global__ __launch_bounds__(256) void kpconv_wmma_f32(
    const float* __restrict__ in_pc,     // (B, IN_PN, CIN)
    const int*   __restrict__ nbr,       // (OUT_PN, M)
    const float* __restrict__ Wt,        // (OUT_PN, M, COUT, CIN)
    const float* __restrict__ bias,      // (OUT_PN, COUT)
    const float* __restrict__ pnb,       // (OUT_PN, M)
    const float* __restrict__ wres,      // (COUT, CIN)
    float*       __restrict__ out)       // (B, OUT_PN, COUT)
{
    __shared__ float lw[2][LDSW_ROWS * LDSW_STRIDE];

    const int p     = blockIdx.x;
    const int tid   = threadIdx.x;
    const int lane  = tid & 31;
    const int wave  = tid >> 5;
    const int l15   = lane & 15;
    const int khalf = lane >> 4;          // 0: K={k0,k0+1}, 1: K={k0+2,k0+3}

    const int b0 = (wave >> 1) * 16;      // batch-tile row of this wave
    const int n0 = (wave & 1) * 32;       // cout base; two tiles at n0, n0+16

    // ---- neighbor ids + normalized residual weights (uniform per block) ----
    int   nid[M_];
    float pm[M_];
    float psum = 0.f;
    #pragma unroll
    for (int m = 0; m < M_; ++m) {
        nid[m] = nbr[p * M_ + m];
        float pv = fabsf(pnb[p * M_ + m]);
        pv = (nid[m] != IN_PN) ? pv : 0.f;
        pm[m] = pv;
        psum += pv;
    }
    const float pinv = 1.f / (psum + 1e-8f);
    #pragma unroll
    for (int m = 0; m < M_; ++m) pm[m] *= pinv;

    // ---- W staging: each thread moves two float4 per neighbor m ----
    const float* wp = Wt + (size_t)p * M_ * COUT * CIN;
    float4 stage0, stage1;
    {
        const float4* src = (const float4*)wp;   // m = 0 chunk (COUT*CIN floats)
        stage0 = src[tid];
        stage1 = src[tid + 256];
    }

    v8f accC0 = {}, accC1 = {};   // conv accumulators (two 16x16 tiles)
    v8f accR0 = {}, accR1 = {};   // residual accumulators
    v2f sfrag[8] = {};            // A-format fragments of S = sum_m pm * A_m

    for (int m = 0; m < M_; ++m) {
        const int buf = m & 1;

        // transposing store: (o, c..c+3) -> packed [c>>1][o][c&1]
        {
            int g = tid;                       // float4 index 0..511
            int o = g >> 3, c4 = g & 7;
            int base = (2 * c4) * LDSW_STRIDE + 2 * o;
            *(v2f*)&lw[buf][base]               = v2f{stage0.x, stage0.y};
            *(v2f*)&lw[buf][base + LDSW_STRIDE] = v2f{stage0.z, stage0.w};
            g = tid + 256;
            o = g >> 3; c4 = g & 7;
            base = (2 * c4) * LDSW_STRIDE + 2 * o;
            *(v2f*)&lw[buf][base]               = v2f{stage1.x, stage1.y};
            *(v2f*)&lw[buf][base + LDSW_STRIDE] = v2f{stage1.z, stage1.w};
        }
        __syncthreads();

        // prefetch next neighbor's weights (global -> regs) to hide latency
        if (m + 1 < M_) {
            const float4* src = (const float4*)(wp + (size_t)(m + 1) * COUT * CIN);
            stage0 = src[tid];
            stage1 = src[tid + 256];
        }

        // A fragments for this m: lane holds batch row b0+l15, channel pair
        v2f af[8];
        if (nid[m] != IN_PN) {   // wave-uniform branch
            const float* row = in_pc + ((size_t)(b0 + l15) * IN_PN + nid[m]) * CIN;
            #pragma unroll
            for (int s = 0; s < 8; ++s)
                af[s] = *(const v2f*)(row + 4 * s + 2 * khalf);
        } else {
            #pragma unroll
            for (int s = 0; s < 8; ++s) af[s] = v2f{0.f, 0.f};
        }

        // residual pre-reduction: S += pm * A  (A-fragment layout)
        #pragma unroll
        for (int s = 0; s < 8; ++s) sfrag[s] += pm[m] * af[s];

        // 8 K-steps of f32 WMMA against both cout tiles
        #pragma unroll
        for (int s = 0; s < 8; ++s) {
            const float* bp = &lw[buf][(2 * s + khalf) * LDSW_STRIDE];
            v2f bf0 = *(const v2f*)(bp + 2 * (n0 + l15));
            v2f bf1 = *(const v2f*)(bp + 2 * (n0 + 16 + l15));
            accC0 = __builtin_amdgcn_wmma_f32_16x16x4_f32(
                false, af[s], false, bf0, (short)0, accC0, false, false);
            accC1 = __builtin_amdgcn_wmma_f32_16x16x4_f32(
                false, af[s], false, bf1, (short)0, accC1, false, false);
        }
    }

    // ---- residual GEMM: accR = S (16x32) x wres^T, K = 32 ----
    #pragma unroll
    for (int s = 0; s < 8; ++s) {
        v2f wf0 = *(const v2f*)(wres + (n0 + l15)      * CIN + 4 * s + 2 * khalf);
        v2f wf1 = *(const v2f*)(wres + (n0 + 16 + l15) * CIN + 4 * s + 2 * khalf);
        accR0 = __builtin_amdgcn_wmma_f32_16x16x4_f32(
            false, sfrag[s], false, wf0, (short)0, accR0, false, false);
        accR1 = __builtin_amdgcn_wmma_f32_16x16x4_f32(
            false, sfrag[s], false, wf1, (short)0, accR1, false, false);
    }

    // ---- epilogue: bias + ELU on conv path, sqrt(.5) blend, store ----
    const float bias0 = bias[p * COUT + n0 + l15];
    const float bias1 = bias[p * COUT + n0 + 16 + l15];
    #pragma unroll
    for (int v = 0; v < 8; ++v) {
        const int b = b0 + v + 8 * khalf;
        float c0 = accC0[v] + bias0;
        float c1 = accC1[v] + bias1;
        c0 = (c0 > 0.f) ? c0 : expm1f(c0);
        c1 = (c1 > 0.f) ? c1 : expm1f(c1);
        const float o0 = SA * c0 + SB * accR0[v];
        const float o1 = SA * c1 + SB * accR1[v];
        float* orow = out + ((size_t)b * OUT_PN + p) * COUT;
        orow[n0 + l15]      = o0;
        orow[n0 + 16 + l15] = o1;
    }
}

extern "C" void kernel_launch(void* const* d_in, const int* in_sizes, int n_in,
                              void* d_out, int out_size, void* d_ws, size_t ws_size,
                              hipStream_t stream) {
    const float* in_pc = (const float*)d_in[0];
    const int*   nbr   = (const int*)  d_in[1];
    const float* Wt    = (const float*)d_in[2];
    const float* bias  = (const float*)d_in[3];
    const float* pnb   = (const float*)d_in[4];
    const float* wres  = (const float*)d_in[5];
    float*       out   = (float*)d_out;

    dim3 grid(OUT_PN);
    dim3 block(256);
    kpconv_wmma_f32<<<grid, block, 0, stream>>>(in_pc, nbr, Wt, bias, pnb, wres, out);
}